// GraphAttent_50036368998988
// MI455X (gfx1250) — compile-verified
//
#include <hip/hip_runtime.h>
#include <hip/hip_bf16.h>

#define IN_F 128
#define OUT_F 128

typedef __attribute__((ext_vector_type(2))) float v2f;
typedef __attribute__((ext_vector_type(8))) float v8f;

// ---------------------------------------------------------------------------
// no-return fp32 global atomic add (STOREcnt path, no CAS loop)
// ---------------------------------------------------------------------------
__device__ __forceinline__ void atomic_add_f32(float* p, float v) {
    asm volatile("global_atomic_add_f32 %0, %1, off"
                 :
                 : "v"(p), "v"(v)
                 : "memory");
}

// ---------------------------------------------------------------------------
// 0) zero output + norm accumulators
// ---------------------------------------------------------------------------
__global__ __launch_bounds__(256) void zero_kernel(float* __restrict__ out, long long n_out,
                                                   float* __restrict__ norm_p,
                                                   float* __restrict__ norm_n, int n_nodes) {
    long long i = (long long)blockIdx.x * blockDim.x + threadIdx.x;
    if (i < n_out) out[i] = 0.0f;
    if (i < n_nodes) {
        norm_p[i] = 0.0f;
        norm_n[i] = 0.0f;
    }
}

// ---------------------------------------------------------------------------
// 1) ptr = X @ W^T  via V_WMMA_F32_16X16X4_F32
//    block = 256 threads (8 waves). grid.x = row tiles, grid.y = side (0/1).
//    Each wave owns one 16x16 output tile; X tile staged in LDS (padded).
// ---------------------------------------------------------------------------
__global__ __launch_bounds__(256) void gemm_ptr_kernel(const float* __restrict__ X,
                                                       const float* __restrict__ Wp,
                                                       const float* __restrict__ Wn,
                                                       float* __restrict__ ptr_p,
                                                       float* __restrict__ ptr_n,
                                                       int n_nodes) {
    __shared__ float xs[16][IN_F + 4];   // +4 pad: 16 rows hit 16 distinct banks

    const int side = blockIdx.y;
    const float* __restrict__ Wm = (side == 0) ? Wp : Wn;
    float* __restrict__ out = (side == 0) ? ptr_p : ptr_n;

    const int m0  = blockIdx.x * 16;
    const int tid = threadIdx.x;

    // cooperative load of the 16x128 X tile (2048 floats / 256 threads)
    for (int i = tid; i < 16 * IN_F; i += 256) {
        int r = i >> 7;
        int c = i & (IN_F - 1);
        int gr = m0 + r;
        xs[r][c] = (gr < n_nodes) ? X[(long long)gr * IN_F + c] : 0.0f;
    }
    __syncthreads();

    const int wave  = tid >> 5;          // 0..7 -> column tile
    const int lane  = tid & 31;
    const int n0    = wave * 16;
    const int lrow  = lane & 15;         // M (for A) / N (for B) within tile
    const int khalf = (lane >> 4) << 1;  // lanes 0-15: K+0,1 ; lanes 16-31: K+2,3

    v8f acc = {};
    #pragma unroll
    for (int k0 = 0; k0 < IN_F; k0 += 4) {
        // A fragment: A[m][k] = X[m0+m][k0+k]
        v2f a;
        a.x = xs[lrow][k0 + khalf + 0];
        a.y = xs[lrow][k0 + khalf + 1];
        // B fragment: B[k][n] = W[n0+n][k0+k]   (B = W^T tile)
        const float* wr = Wm + (long long)(n0 + lrow) * IN_F + k0 + khalf;
        v2f b;
        b.x = wr[0];
        b.y = wr[1];
        acc = __builtin_amdgcn_wmma_f32_16x16x4_f32(false, a, false, b,
                                                    (short)0, acc, false, false);
    }

    // C/D layout: VGPR v -> M = v + (lane>=16 ? 8 : 0), N = lane&15
    const int mbase = (lane >> 4) << 3;  // 0 or 8
    const int ncol  = n0 + (lane & 15);
    #pragma unroll
    for (int v = 0; v < 8; ++v) {
        int gm = m0 + mbase + v;
        if (gm < n_nodes) out[(long long)gm * OUT_F + ncol] = acc[v];
    }
}

// ---------------------------------------------------------------------------
// 2) a_src = ptr @ head[:128], a_dst = ptr @ head[128:]  (both sides)
// ---------------------------------------------------------------------------
__global__ __launch_bounds__(256) void head_kernel(const float* __restrict__ ptr_p,
                                                   const float* __restrict__ ptr_n,
                                                   const float* __restrict__ head_p,
                                                   const float* __restrict__ head_n,
                                                   float* __restrict__ a_src_p,
                                                   float* __restrict__ a_dst_p,
                                                   float* __restrict__ a_src_n,
                                                   float* __restrict__ a_dst_n,
                                                   int n_nodes) {
    int node = blockIdx.x * blockDim.x + threadIdx.x;
    if (node >= n_nodes) return;
    const float4* rp = (const float4*)(ptr_p + (long long)node * OUT_F);
    const float4* rn = (const float4*)(ptr_n + (long long)node * OUT_F);
    const float4* hps = (const float4*)(head_p);
    const float4* hpd = (const float4*)(head_p + OUT_F);
    const float4* hns = (const float4*)(head_n);
    const float4* hnd = (const float4*)(head_n + OUT_F);
    float sp = 0.f, dp = 0.f, sn = 0.f, dn = 0.f;
    #pragma unroll 4
    for (int i = 0; i < OUT_F / 4; ++i) {
        float4 vp = rp[i], vn = rn[i];
        float4 a = hps[i], b = hpd[i], c = hns[i], d = hnd[i];
        sp += vp.x * a.x + vp.y * a.y + vp.z * a.z + vp.w * a.w;
        dp += vp.x * b.x + vp.y * b.y + vp.z * b.z + vp.w * b.w;
        sn += vn.x * c.x + vn.y * c.y + vn.z * c.z + vn.w * c.w;
        dn += vn.x * d.x + vn.y * d.y + vn.z * d.z + vn.w * d.w;
    }
    a_src_p[node] = sp;
    a_dst_p[node] = dp;
    a_src_n[node] = sn;
    a_dst_n[node] = dn;
}

// ---------------------------------------------------------------------------
// 3) per-edge alpha for both signs + atomic norm accumulation
// ---------------------------------------------------------------------------
__global__ __launch_bounds__(256) void edge_alpha_kernel(const long long* __restrict__ eidx,
                                                         const float* __restrict__ ewt,
                                                         const int* __restrict__ esgn,
                                                         const float* __restrict__ a_src_p,
                                                         const float* __restrict__ a_dst_p,
                                                         const float* __restrict__ a_src_n,
                                                         const float* __restrict__ a_dst_n,
                                                         float* __restrict__ alpha_p,
                                                         float* __restrict__ alpha_n,
                                                         float* __restrict__ norm_p,
                                                         float* __restrict__ norm_n,
                                                         int n_edges) {
    int e = blockIdx.x * blockDim.x + threadIdx.x;
    if (e >= n_edges) return;
    long long s = eidx[e];
    long long t = eidx[(long long)n_edges + e];
    int sg = esgn[e];
    float w = ewt[e];

    float xp = a_src_p[s] + a_dst_p[t];
    xp = (xp > 0.f) ? xp : 0.2f * xp;                 // leaky_relu(0.2)
    float ap = __expf(xp) * w * ((sg == 1) ? 1.f : 0.f);

    float xn = a_src_n[s] + a_dst_n[t];
    xn = (xn > 0.f) ? xn : 0.2f * xn;
    float an = __expf(xn) * w * ((sg == -1) ? 1.f : 0.f);

    alpha_p[e] = ap;
    alpha_n[e] = an;
    if (ap != 0.f) atomic_add_f32(&norm_p[t], ap);
    if (an != 0.f) atomic_add_f32(&norm_n[t], an);
}

// ---------------------------------------------------------------------------
// 4) fused message scatter: one wave per edge, float4 per lane,
//    one atomic per output element for BOTH signs combined.
// ---------------------------------------------------------------------------
__global__ __launch_bounds__(256) void message_kernel(const long long* __restrict__ eidx,
                                                      const float* __restrict__ alpha_p,
                                                      const float* __restrict__ alpha_n,
                                                      const float* __restrict__ norm_p,
                                                      const float* __restrict__ norm_n,
                                                      const float* __restrict__ ptr_p,
                                                      const float* __restrict__ ptr_n,
                                                      float* __restrict__ out,
                                                      int n_edges) {
    long long gid = (long long)blockIdx.x * blockDim.x + threadIdx.x;
    int e = (int)(gid >> 5);
    int lane = (int)(gid & 31);
    if (e >= n_edges) return;

    long long s = eidx[e];
    long long t = eidx[(long long)n_edges + e];

    float np = norm_p[t]; np = (np > 0.f) ? np : 1.f;  // jnp.where(norm>0, norm, 1)
    float nn = norm_n[t]; nn = (nn > 0.f) ? nn : 1.f;
    float wp = alpha_p[e] / np;
    float wn = alpha_n[e] / nn;

    const float4 vp = ((const float4*)(ptr_p + s * OUT_F))[lane];
    const float4 vn = ((const float4*)(ptr_n + s * OUT_F))[lane];
    float* o = out + t * OUT_F + lane * 4;
    atomic_add_f32(o + 0, vp.x * wp + vn.x * wn);
    atomic_add_f32(o + 1, vp.y * wp + vn.y * wn);
    atomic_add_f32(o + 2, vp.z * wp + vn.z * wn);
    atomic_add_f32(o + 3, vp.w * wp + vn.w * wn);
}

// ---------------------------------------------------------------------------
extern "C" void kernel_launch(void* const* d_in, const int* in_sizes, int n_in,
                              void* d_out, int out_size, void* d_ws, size_t ws_size,
                              hipStream_t stream) {
    const float*     x    = (const float*)d_in[0];
    const long long* eidx = (const long long*)d_in[1];  // int64 per reference
    const float*     ewt  = (const float*)d_in[2];
    const int*       esgn = (const int*)d_in[3];
    const float*     Wp   = (const float*)d_in[4];
    const float*     Wn   = (const float*)d_in[5];
    const float*     hp   = (const float*)d_in[6];
    const float*     hn   = (const float*)d_in[7];
    float* out = (float*)d_out;

    const int n_nodes = in_sizes[0] / IN_F;
    const int n_edges = in_sizes[2];

    // workspace layout (floats)
    float* ws = (float*)d_ws;
    size_t o = 0;
    float* ptr_p   = ws + o; o += (size_t)n_nodes * OUT_F;
    float* ptr_n   = ws + o; o += (size_t)n_nodes * OUT_F;
    float* a_src_p = ws + o; o += n_nodes;
    float* a_dst_p = ws + o; o += n_nodes;
    float* a_src_n = ws + o; o += n_nodes;
    float* a_dst_n = ws + o; o += n_nodes;
    float* norm_p  = ws + o; o += n_nodes;
    float* norm_n  = ws + o; o += n_nodes;
    float* alpha_p = ws + o; o += n_edges;
    float* alpha_n = ws + o; o += n_edges;

    const long long n_out = (long long)n_nodes * OUT_F;

    // 0) zero accumulators
    {
        long long total = n_out;
        int blocks = (int)((total + 255) / 256);
        zero_kernel<<<blocks, 256, 0, stream>>>(out, n_out, norm_p, norm_n, n_nodes);
    }
    // 1) WMMA GEMM (both sides via grid.y)
    {
        dim3 grid((n_nodes + 15) / 16, 2);
        gemm_ptr_kernel<<<grid, 256, 0, stream>>>(x, Wp, Wn, ptr_p, ptr_n, n_nodes);
    }
    // 2) head projections
    head_kernel<<<(n_nodes + 255) / 256, 256, 0, stream>>>(
        ptr_p, ptr_n, hp, hn, a_src_p, a_dst_p, a_src_n, a_dst_n, n_nodes);
    // 3) edge alphas + norms
    edge_alpha_kernel<<<(n_edges + 255) / 256, 256, 0, stream>>>(
        eidx, ewt, esgn, a_src_p, a_dst_p, a_src_n, a_dst_n,
        alpha_p, alpha_n, norm_p, norm_n, n_edges);
    // 4) fused message scatter (wave per edge)
    {
        long long threads = (long long)n_edges * 32;
        int blocks = (int)((threads + 255) / 256);
        message_kernel<<<blocks, 256, 0, stream>>>(
            eidx, alpha_p, alpha_n, norm_p, norm_n, ptr_p, ptr_n, out, n_edges);
    }
}